// MGRU_61323543052768
// MI455X (gfx1250) — compile-verified
//
#include <hip/hip_runtime.h>
#include <cmath>

// Problem constants (match the reference)
#define BB  64
#define TT  512
#define EIc 1024
#define EOc 1024

typedef float v2f __attribute__((ext_vector_type(2)));
typedef float v8f __attribute__((ext_vector_type(8)));

__device__ __forceinline__ v8f wmma4(v2f a, v2f b, v8f c) {
  // D = A(16x4,f32) * B(4x16,f32) + C(16x16,f32)
  return __builtin_amdgcn_wmma_f32_16x16x4_f32(
      /*neg_a=*/false, a, /*neg_b=*/false, b,
      /*c_mod=*/(short)0, c, /*reuse_a=*/false, /*reuse_b=*/false);
}

// -----------------------------------------------------------------------------
// Repack W[K,N] (row-major) into fragment-contiguous layout:
//   fragment (k4, n16) = 16x4 B-tile for WMMA; block id = k4*(N/16) + n16,
//   each block is 32 lanes x 2 floats (8B per lane, contiguous 256B).
//   lane l (half = l>>4, n = l&15): pair { W[4*k4+2*half][16*n16+n],
//                                          W[4*k4+2*half+1][16*n16+n] }
// -----------------------------------------------------------------------------
__global__ __launch_bounds__(256) void repack_w(
    const float* __restrict__ W, float* __restrict__ Wp)
{
  const int tid  = blockIdx.x * 256 + threadIdx.x;   // one thread per lane-slot
  const int lane = tid & 31;
  const int blk  = tid >> 5;                         // 0 .. (K/4)*(N/16)-1
  const int k4   = blk >> 6;                         // K/4 rows, N/16 = 64
  const int n16  = blk & 63;
  const int half = lane >> 4;
  const int l16  = lane & 15;
  const int k = 4 * k4 + 2 * half;
  const int n = 16 * n16 + l16;
  v2f v;
  v.x = W[(size_t)k * EOc + n];
  v.y = W[(size_t)(k + 1) * EOc + n];
  *(v2f*)(Wp + (size_t)blk * 64 + lane * 2) = v;
}

// -----------------------------------------------------------------------------
// Phase 1: G = tanh(X @ Wxh), using packed Wxh.
//   X: [M=32768, K=1024], Wp: packed, G: [M, N=1024] (outs region of d_out).
// One wave computes a 32x64 strip: 2 A frags x 4 B frags -> 8 WMMA per K-step,
// with 2 global b64 (A) + 4 global b64 (packed B) loads.
// -----------------------------------------------------------------------------
__global__ __launch_bounds__(256) void mgru_gemm_tanh(
    const float* __restrict__ X, const float* __restrict__ Wp,
    float* __restrict__ G)
{
  const int lane = threadIdx.x & 31;
  const int wave = threadIdx.x >> 5;
  const int gw   = blockIdx.x * 8 + wave;
  const int colStrips = EOc / 64;             // 16
  const int rowBase = (gw / colStrips) * 32;
  const int colStrip = gw % colStrips;
  const int colBase = colStrip * 64;
  const int half = lane >> 4;
  const int l16  = lane & 15;

  v8f acc[8];
#pragma unroll
  for (int i = 0; i < 8; ++i) acc[i] = v8f{};

  const float* a0 = X + (size_t)(rowBase + l16) * EIc + 2 * half;
  const float* a1 = a0 + (size_t)16 * EIc;
  // packed B base for this column strip: blocks (k4, colStrip*4 + j)
  const float* bp = Wp + (size_t)colStrip * 4 * 64 + lane * 2;

  for (int k0 = 0; k0 < EIc; k0 += 4) {
    v2f A0 = *(const v2f*)(a0 + k0);
    v2f A1 = *(const v2f*)(a1 + k0);
    const float* bb = bp + (size_t)(k0 >> 2) * (64 * 64);  // 64 blocks per k4
    v2f B0 = *(const v2f*)(bb + 0);
    v2f B1 = *(const v2f*)(bb + 64);
    v2f B2 = *(const v2f*)(bb + 128);
    v2f B3 = *(const v2f*)(bb + 192);
    acc[0] = wmma4(A0, B0, acc[0]);
    acc[1] = wmma4(A0, B1, acc[1]);
    acc[2] = wmma4(A0, B2, acc[2]);
    acc[3] = wmma4(A0, B3, acc[3]);
    acc[4] = wmma4(A1, B0, acc[4]);
    acc[5] = wmma4(A1, B1, acc[5]);
    acc[6] = wmma4(A1, B2, acc[6]);
    acc[7] = wmma4(A1, B3, acc[7]);
  }

#pragma unroll
  for (int mi = 0; mi < 2; ++mi) {
    float* gout = G + (size_t)(rowBase + 16 * mi + 8 * half) * EOc + colBase + l16;
#pragma unroll
    for (int r = 0; r < 8; ++r) {
#pragma unroll
      for (int j = 0; j < 4; ++j) {
        gout[(size_t)r * EOc + 16 * j] = tanhf(acc[mi * 4 + j][r]);
      }
    }
  }
}

// -----------------------------------------------------------------------------
// Phase 2: persistent recurrent scan.
//   16 workgroups x 128 threads (4 waves). WG g owns column strip g (64 cols).
//   Prologue: stage whf[:, 64g:64g+64] into LDS (256KB) in packed fragment
//   layout. Then loop t = 0..T-1:
//     each wave w computes 16x64 tile of F = sigmoid(Hold @ whf) for rows
//     16w..16w+15, B frags from LDS (ds_load_b64), A frags from Hold (global).
//     update: h = (1-f)*Hold + f*G; write Outs[:,t,:] (over gt) and Hnew;
//     t == T-1 also writes hT. Steps separated by a monotonic global atomic
//     spin barrier across the 16 resident WGs.
// -----------------------------------------------------------------------------
__global__ __launch_bounds__(128) void mgru_scan(
    const float* __restrict__ Whf, float* __restrict__ Outs,
    float* __restrict__ Ha, float* __restrict__ Hb,
    float* __restrict__ hT, unsigned int* __restrict__ ctr)
{
  extern __shared__ float lds[];              // 1024 blocks * 64 floats = 256KB

  const int lane = threadIdx.x & 31;
  const int wave = threadIdx.x >> 5;          // 0..3 -> row tile
  const int half = lane >> 4;
  const int l16  = lane & 15;
  const int colStrip = blockIdx.x;            // 0..15
  const int colBase  = colStrip * 64;

  // ---- Stage whf column slice into LDS, packed fragment layout ----
  // block blk = k4*4 + sub (sub = 16-col subtile inside this strip)
  for (int blk = wave; blk < 1024; blk += 4) {
    const int k4  = blk >> 2;
    const int sub = blk & 3;
    const int k = 4 * k4 + 2 * half;
    const int n = colBase + 16 * sub + l16;
    v2f v;
    v.x = Whf[(size_t)k * EOc + n];
    v.y = Whf[(size_t)(k + 1) * EOc + n];
    *(v2f*)(lds + (size_t)blk * 64 + lane * 2) = v;
  }
  __syncthreads();

  const int rowBase = wave * 16;
  const float* lb = lds + lane * 2;

  for (int t = 0; t < TT; ++t) {
    const float* Hold = (t & 1) ? Hb : Ha;
    float*       Hnew = (t & 1) ? Ha : Hb;

    v8f acc[4];
#pragma unroll
    for (int i = 0; i < 4; ++i) acc[i] = v8f{};

    const float* arow = Hold + (size_t)(rowBase + l16) * EOc + 2 * half;
    for (int k0 = 0; k0 < EOc; k0 += 4) {
      v2f a = *(const v2f*)(arow + k0);
      const float* bb = lb + (size_t)(k0 >> 2) * 256;   // 4 blocks per k4
      v2f b0 = *(const v2f*)(bb + 0);
      v2f b1 = *(const v2f*)(bb + 64);
      v2f b2 = *(const v2f*)(bb + 128);
      v2f b3 = *(const v2f*)(bb + 192);
      acc[0] = wmma4(a, b0, acc[0]);
      acc[1] = wmma4(a, b1, acc[1]);
      acc[2] = wmma4(a, b2, acc[2]);
      acc[3] = wmma4(a, b3, acc[3]);
    }

    const int row0 = rowBase + 8 * half;
#pragma unroll
    for (int r = 0; r < 8; ++r) {
      const int b = row0 + r;
      const float* hop = Hold + (size_t)b * EOc + colBase + l16;
      float*       op  = Outs + ((size_t)b * TT + t) * EOc + colBase + l16;
      float*       hnp = Hnew + (size_t)b * EOc + colBase + l16;
#pragma unroll
      for (int j = 0; j < 4; ++j) {
        const float f    = 1.0f / (1.0f + expf(-acc[j][r]));
        const float hold = hop[16 * j];
        const float g    = op[16 * j];            // gt from phase 1
        const float hn   = (1.0f - f) * hold + f * g;
        op[16 * j]  = hn;                         // out_t = h_t
        hnp[16 * j] = hn;
        if (t == TT - 1) hT[(size_t)b * EOc + colBase + l16 + 16 * j] = hn;
      }
    }

    // ---- inter-step barrier across the 16 workgroups ----
    __threadfence();
    __syncthreads();
    if (threadIdx.x == 0) {
      __hip_atomic_fetch_add(ctr, 1u, __ATOMIC_ACQ_REL, __HIP_MEMORY_SCOPE_AGENT);
      const unsigned int target = 16u * (unsigned int)(t + 1);
      while (__hip_atomic_load(ctr, __ATOMIC_ACQUIRE, __HIP_MEMORY_SCOPE_AGENT) < target) {
        __builtin_amdgcn_s_sleep(1);
      }
    }
    __syncthreads();
  }
}

// -----------------------------------------------------------------------------
extern "C" void kernel_launch(void* const* d_in, const int* in_sizes, int n_in,
                              void* d_out, int out_size, void* d_ws, size_t ws_size,
                              hipStream_t stream) {
  const float* x   = (const float*)d_in[0];   // [B,T,EI]
  const float* h0  = (const float*)d_in[1];   // [B,EO]
  const float* wxh = (const float*)d_in[2];   // [EI,EO]
  const float* whf = (const float*)d_in[3];   // [EO,EO]

  float* outs = (float*)d_out;                     // [B,T,EO]
  float* hT   = outs + (size_t)BB * TT * EOc;      // [B,EO]

  // Workspace layout
  float* wxhP = (float*)d_ws;                              // 4MB packed wxh
  float* Ha   = wxhP + (size_t)EIc * EOc;                  // 256KB
  float* Hb   = Ha + (size_t)BB * EOc;                     // 256KB
  unsigned int* ctr = (unsigned int*)(Hb + (size_t)BB * EOc);

  // Zero the barrier counter (graph-capture safe memset node).
  hipMemsetAsync(ctr, 0, 128, stream);

  // Repack wxh into fragment-contiguous layout.
  repack_w<<<(EIc / 4) * (EOc / 16) * 32 / 256, 256, 0, stream>>>(wxh, wxhP);

  // Phase 1: G = tanh(X @ wxh) into the outs region.
  const int p1Waves = (BB * TT / 32) * (EOc / 64);   // 16384
  mgru_gemm_tanh<<<p1Waves / 8, 256, 0, stream>>>(x, wxhP, outs);

  // Seed H with h0.
  hipMemcpyAsync(Ha, h0, sizeof(float) * BB * EOc, hipMemcpyDeviceToDevice, stream);

  // Phase 2: persistent scan, 16 WGs x 128 threads, 256KB dynamic LDS each.
  hipFuncSetAttribute((const void*)mgru_scan,
                      hipFuncAttributeMaxDynamicSharedMemorySize, 256 * 1024);
  mgru_scan<<<16, 128, 256 * 1024, stream>>>(whf, outs, Ha, Hb, hT, ctr);
}